// BGNLLLoss_77068893160394
// MI455X (gfx1250) — compile-verified
//
#include <hip/hip_runtime.h>
#include <hip/hip_bf16.h>

// Bivariate-Gaussian NLL mean over T*N = 4096*4096 f32 elements.
// Bandwidth-bound: 448 MB streamed once -> ~19us roofline at 23.3 TB/s.
// Kernel 1: vectorized NT streaming + deterministic block partial sums.
// Kernel 2: single-wave final reduction using V_WMMA_F32_16X16X4_F32
//           (A = lane values, B = ones -> D rows = v[m]+v[m+16]).

typedef float  v2f  __attribute__((ext_vector_type(2)));
typedef float  v4f  __attribute__((ext_vector_type(4)));
typedef float  v8f  __attribute__((ext_vector_type(8)));

#define TT 4096
#define NN 4096
#define M_ELEMS (TT * NN)          // 16,777,216
#define NUM_BLOCKS 2048
#define BLOCK 256
#define LOG_2PI  1.8378770664093453f
#define HALF_LN2 0.34657359027997264f   // 0.5 * ln(2)
#define NLL_MAX  46.051701859880914f    // -log(1e-20)

__global__ __launch_bounds__(BLOCK)
void bgnll_partial_kernel(const float* __restrict__ targets,
                          const float* __restrict__ mux,
                          const float* __restrict__ muy,
                          const float* __restrict__ sx,
                          const float* __restrict__ sy,
                          const float* __restrict__ corr,
                          float* __restrict__ partials) {
    const int nGroups = M_ELEMS / 4;                  // float4 groups
    const int tid     = blockIdx.x * BLOCK + threadIdx.x;
    const int stride  = NUM_BLOCKS * BLOCK;

    const v4f* __restrict__ mux4  = (const v4f*)mux;
    const v4f* __restrict__ muy4  = (const v4f*)muy;
    const v4f* __restrict__ sx4   = (const v4f*)sx;
    const v4f* __restrict__ sy4   = (const v4f*)sy;
    const v4f* __restrict__ cr4   = (const v4f*)corr;
    const v4f* __restrict__ tg4   = (const v4f*)targets;

    float acc = 0.0f;
    for (int i = tid; i < nGroups; i += stride) {
        // Streamed once; 448MB >> 192MB L2 -> non-temporal B128 loads.
        v4f mx = __builtin_nontemporal_load(mux4 + i);
        v4f my = __builtin_nontemporal_load(muy4 + i);
        v4f vx = __builtin_nontemporal_load(sx4  + i);
        v4f vy = __builtin_nontemporal_load(sy4  + i);
        v4f cr = __builtin_nontemporal_load(cr4  + i);
        // targets interleaved (x,y): group i covers floats [8i, 8i+8)
        v4f t0 = __builtin_nontemporal_load(tg4 + 2 * i);
        v4f t1 = __builtin_nontemporal_load(tg4 + 2 * i + 1);
        v4f X = {t0.x, t0.z, t1.x, t1.z};
        v4f Y = {t0.y, t0.w, t1.y, t1.w};

        #pragma unroll
        for (int j = 0; j < 4; ++j) {
            float nx = X[j] - mx[j];
            float ny = Y[j] - my[j];
            float rsx = __builtin_amdgcn_rcpf(vx[j]);
            float rsy = __builtin_amdgcn_rcpf(vy[j]);
            float ax = nx * rsx;
            float ay = ny * rsy;
            float c  = cr[j];
            float z  = ax * ax + ay * ay - 2.0f * c * ax * ay;
            float nr = 1.0f - c * c;                       // 1 - rho^2 > 0
            float sxsy = vx[j] * vy[j];
            // -log(pdf) = z/(2nr) + log(2pi) + 0.5*log(sxsy^2 * nr)
            float nll = z * (0.5f * __builtin_amdgcn_rcpf(nr))
                      + LOG_2PI
                      + HALF_LN2 * __builtin_amdgcn_logf(sxsy * sxsy * nr);
            acc += fminf(nll, NLL_MAX);                    // clip(pdf, 1e-20)
        }
    }

    // wave32 reduction
    #pragma unroll
    for (int off = 16; off > 0; off >>= 1)
        acc += __shfl_down(acc, off, 32);

    __shared__ float wsum[BLOCK / 32];
    const int lane = threadIdx.x & 31;
    const int wave = threadIdx.x >> 5;
    if (lane == 0) wsum[wave] = acc;
    __syncthreads();

    if (wave == 0) {
        float s = (lane < BLOCK / 32) ? wsum[lane] : 0.0f;
        #pragma unroll
        for (int off = 4; off > 0; off >>= 1)
            s += __shfl_down(s, off, 32);
        if (lane == 0) partials[blockIdx.x] = s;
    }
}

// Single wave (32 threads): EXEC all-ones as WMMA requires.
__global__ __launch_bounds__(32)
void bgnll_final_kernel(const float* __restrict__ partials,
                        float* __restrict__ out) {
    const int lane = threadIdx.x;      // 0..31, no divergence before WMMA
    float v = 0.0f;
    for (int i = lane; i < NUM_BLOCKS; i += 32)
        v += partials[i];

    // A (16x4 f32): a.x -> lanes 0-15: A[m=lane][K=0]; lanes 16-31: A[m][K=2]
    // => row m of A = {v[m], 0, v[m+16], 0}.  B = ones(4x16).
    // D[m][n] = v[m] + v[m+16]; sum of column 0 over all 16 rows = total.
    v2f a = {v, 0.0f};
    v2f b = {1.0f, 1.0f};
    v8f c = {};
    c = __builtin_amdgcn_wmma_f32_16x16x4_f32(
            /*neg_a=*/false, a, /*neg_b=*/false, b,
            /*c_mod=*/(short)0, c, /*reuse_a=*/false, /*reuse_b=*/false);

    // lane 0 holds D[0..7][0], lane 16 holds D[8..15][0]
    float s = c[0] + c[1] + c[2] + c[3] + c[4] + c[5] + c[6] + c[7];
    float hi = __shfl_down(s, 16, 32);
    if (lane == 0)
        out[0] = (s + hi) * (1.0f / (float)M_ELEMS);   // exact 2^-24 scale
}

extern "C" void kernel_launch(void* const* d_in, const int* in_sizes, int n_in,
                              void* d_out, int out_size, void* d_ws, size_t ws_size,
                              hipStream_t stream) {
    const float* targets = (const float*)d_in[0];
    const float* mux     = (const float*)d_in[1];
    const float* muy     = (const float*)d_in[2];
    const float* sx      = (const float*)d_in[3];
    const float* sy      = (const float*)d_in[4];
    const float* corr    = (const float*)d_in[5];
    float* out      = (float*)d_out;
    float* partials = (float*)d_ws;     // NUM_BLOCKS floats, fully overwritten

    bgnll_partial_kernel<<<NUM_BLOCKS, BLOCK, 0, stream>>>(
        targets, mux, muy, sx, sy, corr, partials);
    bgnll_final_kernel<<<1, 32, 0, stream>>>(partials, out);
}